// CRPSSpectralLoss_68556267979183
// MI455X (gfx1250) — compile-verified
//
#include <hip/hip_runtime.h>
#include <math.h>
#include <stdint.h>

typedef __attribute__((ext_vector_type(2))) float v2f;
typedef __attribute__((ext_vector_type(8))) float v8f;

#define B_   8
#define M_   16
#define C_   3
#define H_   128
#define W_   128
#define G_   (H_ * W_)              // 16384
#define GF_  (H_ * (W_ / 2 + 1))    // 8320 (full rfft2 grid; mask zeros outside)
#define CUT_H 32
#define CUT_W 16
#define NBIN (CUT_H * CUT_W)        // 512 nonzero low-pass bins

// d_ws float layout: [0]=S1_pointwise [1]=S2_pointwise [2]=S1_freq [3]=S2_freq

__global__ void zero_acc_kernel(float* ws) {
    if (threadIdx.x < 4) ws[threadIdx.x] = 0.0f;
}

// ---------------------------------------------------------------------------
// Pointwise CRPS sums over the raw spatial fields.
// One thread per pixel (b, c, g). S1 = sum_m |p_m - t|, S2 = sum_{i<j} |p_i-p_j|.
// ---------------------------------------------------------------------------
__global__ void __launch_bounds__(256)
crps_pointwise_kernel(const float* __restrict__ tgt,
                      const float* __restrict__ out,
                      float* __restrict__ ws) {
    __shared__ float red[512];
    const int tid = threadIdx.x;
    const int P   = blockIdx.x * 256 + tid;
    const int b   = P / (C_ * G_);
    const int rem = P % (C_ * G_);
    const int c   = rem / G_;
    const int g   = rem % G_;

    const float t = tgt[(b * C_ + c) * G_ + g];

    // hint the streaming ensemble reads into cache (global_prefetch_b8)
    __builtin_prefetch(out + ((b * M_) * C_ + c) * G_ + g, 0, 1);

    float p[M_];
#pragma unroll
    for (int m = 0; m < M_; ++m)
        p[m] = out[((b * M_ + m) * C_ + c) * G_ + g];

    float s1 = 0.0f, s2 = 0.0f;
#pragma unroll
    for (int m = 0; m < M_; ++m) s1 += fabsf(p[m] - t);
#pragma unroll
    for (int i = 0; i < M_; ++i)
#pragma unroll
        for (int j = i + 1; j < M_; ++j) s2 += fabsf(p[i] - p[j]);

    red[tid] = s1;
    red[256 + tid] = s2;
    __syncthreads();
    for (int s = 128; s > 0; s >>= 1) {
        if (tid < s) {
            red[tid] += red[tid + s];
            red[256 + tid] += red[256 + tid + s];
        }
        __syncthreads();
    }
    if (tid == 0) {
        atomicAdd(&ws[0], red[0]);
        atomicAdd(&ws[1], red[256]);
    }
}

// ---------------------------------------------------------------------------
// Spectral CRPS: per (b,c), compute the 32x16 low-pass DFT magnitudes of the
// 16 ensemble members + target via V_WMMA_F32_16X16X4_F32, then pairwise CRPS
// over those bins in LDS.
//
// Images are streamed into LDS with double-buffered coalesced
// GLOBAL_LOAD_ASYNC_TO_LDS_B128 (ASYNCcnt), overlapping the next member's
// transfer with the current member's WMMA DFT.
//
// Stage 1: Y(128x32) = X(128x128) * Wt(128x32), Wt = [cos | -sin]  (per kw)
// Stage 2: Z(64x32)  = Ft(64x128) * Y(128x32),  Ft = [Ch ; Sh]    (per kh)
//          Re = Z[kh,kw] + Z[32+kh,16+kw];  Im = Z[kh,16+kw] - Z[32+kh,kw]
// ---------------------------------------------------------------------------
__global__ void __launch_bounds__(256)
spectral_crps_kernel(const float* __restrict__ tgt,
                     const float* __restrict__ out,
                     float* __restrict__ ws) {
    extern __shared__ float lds[];
    float* Wt   = lds;                 // 128*32 = 4096 f
    float* Ft   = Wt + 128 * 32;       // 64*128 = 8192 f
    float* Y    = Ft + 64 * 128;       // 128*32 = 4096 f
    float* Z    = Y + 128 * 32;        // 64*32  = 2048 f
    float* Mg   = Z + 64 * 32;         // 17*512 = 8704 f
    float* red  = Mg + 17 * NBIN;      // 512 f
    float* img0 = red + 512;           // 16384 f (64 KB)
    float* img1 = img0 + G_;           // 16384 f (64 KB)

    const int tid  = threadIdx.x;
    const int wid  = tid >> 5;        // wave id 0..7 (wave32)
    const int lane = tid & 31;
    const int half = lane >> 4;       // 0: lanes 0-15, 1: lanes 16-31
    const int mrow = lane & 15;

    const int bc = blockIdx.x;
    const int b  = bc / C_;
    const int c  = bc % C_;

    // Async stage of one 128x128 fp32 image into an LDS buffer:
    // 4096 x b128 chunks, 16 per lane, fully coalesced.
    auto issue_image_async = [&](int im, float* dstbuf) {
        const float* gimg = (im < M_)
            ? out + ((b * M_ + im) * C_ + c) * G_
            : tgt + (b * C_ + c) * G_;
        const uint32_t ldsbase = (uint32_t)(uintptr_t)dstbuf;
#pragma unroll
        for (int i = 0; i < 16; ++i) {
            const int q = tid + i * 256;                       // chunk id < 4096
            const uint64_t ga = (uint64_t)(uintptr_t)(gimg + q * 4);
            const uint32_t la = ldsbase + (uint32_t)(q * 16);
            asm volatile("global_load_async_to_lds_b128 %0, %1, off"
                         :: "v"(la), "v"(ga) : "memory");
        }
    };

    // Kick off image 0 before the twiddle build so the transfer overlaps
    // the transcendental-heavy setup.
    issue_image_async(0, img0);

    const float OMEGA = 6.28318530717958647692f / 128.0f;  // 2*pi/128

    for (int i = tid; i < 128 * 32; i += 256) {
        const int wv = i >> 5, kw = i & 31;
        const float th = OMEGA * (float)(wv * (kw & 15));
        Wt[i] = (kw < 16) ? cosf(th) : -sinf(th);
    }
    for (int i = tid; i < 64 * 128; i += 256) {
        const int r = i >> 7, h = i & 127;
        const float th = OMEGA * (float)(h * (r & 31));
        Ft[i] = (r < 32) ? cosf(th) : sinf(th);
    }

    for (int im = 0; im <= M_; ++im) {
        float* imgbuf = (im & 1) ? img1 : img0;
        if (im < M_) {
            // Prefetch next image into the other buffer, then wait until only
            // the 16 just-issued async ops remain outstanding (async loads
            // complete in order -> current image is fully in LDS).
            issue_image_async(im + 1, (im & 1) ? img0 : img1);
            asm volatile("s_wait_asynccnt 0x10" ::: "memory");
        } else {
            asm volatile("s_wait_asynccnt 0x0" ::: "memory");
        }
        __syncthreads();  // image (and, on first pass, twiddles) visible to all waves

        // ----- Stage 1: Y = X * Wt (8 M-tiles x 2 N-tiles, K=128) -----
        for (int job = wid; job < 16; job += 8) {
            const int mt = job & 7;
            const int nt = job >> 3;
            v8f acc = {};
            for (int k = 0; k < 32; ++k) {
                const int acol = k * 4 + half * 2;
                const int arow = mt * 16 + mrow;
                v2f a, bb;
                a.x  = imgbuf[arow * W_ + acol];
                a.y  = imgbuf[arow * W_ + acol + 1];
                bb.x = Wt[(acol + 0) * 32 + nt * 16 + mrow];
                bb.y = Wt[(acol + 1) * 32 + nt * 16 + mrow];
                acc = __builtin_amdgcn_wmma_f32_16x16x4_f32(
                    false, a, false, bb, (short)0, acc, false, false);
            }
#pragma unroll
            for (int v = 0; v < 8; ++v)
                Y[(mt * 16 + v + half * 8) * 32 + nt * 16 + mrow] = acc[v];
        }
        __syncthreads();

        // ----- Stage 2: Z = Ft * Y (4 M-tiles x 2 N-tiles, K=128) -----
        {
            const int mt = wid & 3;
            const int nt = wid >> 2;
            v8f acc = {};
            for (int k = 0; k < 32; ++k) {
                const int acol = k * 4 + half * 2;
                v2f a, bb;
                a.x  = Ft[(mt * 16 + mrow) * 128 + acol];
                a.y  = Ft[(mt * 16 + mrow) * 128 + acol + 1];
                bb.x = Y[(acol + 0) * 32 + nt * 16 + mrow];
                bb.y = Y[(acol + 1) * 32 + nt * 16 + mrow];
                acc = __builtin_amdgcn_wmma_f32_16x16x4_f32(
                    false, a, false, bb, (short)0, acc, false, false);
            }
#pragma unroll
            for (int v = 0; v < 8; ++v)
                Z[(mt * 16 + v + half * 8) * 32 + nt * 16 + mrow] = acc[v];
        }
        __syncthreads();

        // ----- Magnitudes of the 32x16 low-pass corner -----
        for (int i = tid; i < NBIN; i += 256) {
            const int kh = i >> 4, kw = i & 15;
            const float xr = Z[kh * 32 + kw]      + Z[(32 + kh) * 32 + 16 + kw];
            const float xi = Z[kh * 32 + 16 + kw] - Z[(32 + kh) * 32 + kw];
            Mg[im * NBIN + i] = sqrtf(xr * xr + xi * xi);
        }
        __syncthreads();  // protects Z/Y and the stage-1 image buffer for reuse
    }

    // ----- Pairwise CRPS over the 512 nonzero frequency bins -----
    float s1 = 0.0f, s2 = 0.0f;
    for (int g = tid; g < NBIN; g += 256) {
        const float t = Mg[M_ * NBIN + g];
        float p[M_];
#pragma unroll
        for (int m = 0; m < M_; ++m) p[m] = Mg[m * NBIN + g];
#pragma unroll
        for (int m = 0; m < M_; ++m) s1 += fabsf(p[m] - t);
#pragma unroll
        for (int i = 0; i < M_; ++i)
#pragma unroll
            for (int j = i + 1; j < M_; ++j) s2 += fabsf(p[i] - p[j]);
    }
    red[tid] = s1;
    red[256 + tid] = s2;
    __syncthreads();
    for (int s = 128; s > 0; s >>= 1) {
        if (tid < s) {
            red[tid] += red[tid + s];
            red[256 + tid] += red[256 + tid + s];
        }
        __syncthreads();
    }
    if (tid == 0) {
        atomicAdd(&ws[2], red[0]);
        atomicAdd(&ws[3], red[256]);
    }
}

// ---------------------------------------------------------------------------
// Combine the four accumulated sums with the reference's exact denominators.
// ---------------------------------------------------------------------------
__global__ void finalize_kernel(const float* __restrict__ ws,
                                float* __restrict__ outp) {
    if (threadIdx.x == 0 && blockIdx.x == 0) {
        const double eps = 0.05 / (double)M_;
        const double dp  = (double)B_ * M_ * C_ * G_;
        const double df  = (double)B_ * M_ * C_ * GF_;

        const double t1p = (double)ws[0] / dp;
        const double t2p = (2.0 * (double)ws[1] / (double)(M_ - 1)) / dp * (1.0 - eps);
        const double crps_p = t1p - 0.5 * t2p;

        const double t1f = (double)ws[2] / df;
        const double t2f = (2.0 * (double)ws[3] / (double)(M_ - 1)) / df * (1.0 - eps);
        const double crps_f = t1f - 0.5 * t2f;

        outp[0] = (float)(crps_p + 0.1 * crps_f);
    }
}

extern "C" void kernel_launch(void* const* d_in, const int* in_sizes, int n_in,
                              void* d_out, int out_size, void* d_ws, size_t ws_size,
                              hipStream_t stream) {
    const float* tgt = (const float*)d_in[0];  // (8, 3, 128, 128)
    const float* out = (const float*)d_in[1];  // (8, 16, 3, 128, 128)
    float* ws = (float*)d_ws;
    float* o  = (float*)d_out;

    hipLaunchKernelGGL(zero_acc_kernel, dim3(1), dim3(32), 0, stream, ws);

    const int npix = B_ * C_ * G_;  // 393216
    hipLaunchKernelGGL(crps_pointwise_kernel, dim3(npix / 256), dim3(256), 0,
                       stream, tgt, out, ws);

    const size_t shmem =
        (size_t)(128 * 32 + 64 * 128 + 128 * 32 + 64 * 32 + 17 * NBIN + 512 +
                 2 * G_) * sizeof(float);  // 241664 bytes
    hipLaunchKernelGGL(spectral_crps_kernel, dim3(B_ * C_), dim3(256), shmem,
                       stream, tgt, out, ws);

    hipLaunchKernelGGL(finalize_kernel, dim3(1), dim3(1), 0, stream, ws, o);
}